// GAT_27797028340068
// MI455X (gfx1250) — compile-verified
//
#include <hip/hip_runtime.h>

typedef __attribute__((ext_vector_type(2))) float v2f;
typedef __attribute__((ext_vector_type(8))) float v8f;

// ---------------------------------------------------------------------------
// Ordered-uint encoding so unsigned atomicMax == float max (handles negatives)
// ---------------------------------------------------------------------------
__device__ __forceinline__ unsigned f2ord(float f) {
    unsigned u = __float_as_uint(f);
    return ((int)u >= 0) ? (u | 0x80000000u) : ~u;
}
__device__ __forceinline__ float ord2f(unsigned k) {
    unsigned b = ((int)k < 0) ? (k & 0x7FFFFFFFu) : ~k;
    return __uint_as_float(b);
}

// ---------------------------------------------------------------------------
// WMMA f32 GEMM: C[N x 16*MT] = A[N x K] * B[K x 16*MT]
// One wave computes a 16-row stripe across the full output width (MT tiles),
// so A is streamed from HBM exactly once. Emits v_wmma_f32_16x16x4_f32 and
// global_prefetch_b8 (per-lane 256B-ahead prefetch of the A row stream).
// A-frag (16x4 f32): lanes 0-15 rows, VGPRs = K {0,1}; lanes 16-31 K {2,3}.
// B-frag (4x16 f32): lanes 0-15 cols, rows K {0,1}; lanes 16-31 rows K {2,3}.
// ---------------------------------------------------------------------------
template <int MT>
__global__ void gemm_wmma_f32(const float* __restrict__ A,
                              const float* __restrict__ B,
                              float* __restrict__ C, int N, int K) {
    const int M = 16 * MT;
    int wave = (blockIdx.x * blockDim.x + threadIdx.x) >> 5;
    int lane = threadIdx.x & 31;
    int ntiles = N >> 4;                 // N multiple of 16 (100000 = 6250*16)
    if (wave >= ntiles) return;
    int row0  = wave * 16;
    int khalf = (lane >> 4) * 2;         // 0 for lanes 0-15, 2 for lanes 16-31
    int r     = row0 + (lane & 15);
    int col   = lane & 15;

    v8f acc[MT] = {};
    const float* arow = A + (size_t)r * K + khalf;
    for (int k = 0; k < K; k += 16) {
        // speculative prefetch 256B ahead in this lane's contiguous row stream
        __builtin_prefetch(arow + k + 64, 0, 1);
#pragma unroll
        for (int kk = 0; kk < 16; kk += 4) {
            v2f a;
            a.x = arow[k + kk];
            a.y = arow[k + kk + 1];
            const float* bp = B + (size_t)(k + kk + khalf) * M + col;
#pragma unroll
            for (int t = 0; t < MT; t++) {
                v2f b;
                b.x = bp[t * 16];
                b.y = bp[M + t * 16];
                acc[t] = __builtin_amdgcn_wmma_f32_16x16x4_f32(
                    false, a, false, b, (short)0, acc[t], false, false);
            }
        }
    }
    // C/D layout: VGPR i -> row (i + 0) for lanes 0-15, row (i + 8) for 16-31
    int rowadd = (lane >> 4) * 8;
#pragma unroll
    for (int t = 0; t < MT; t++)
#pragma unroll
        for (int i = 0; i < 8; i++)
            C[(size_t)(row0 + rowadd + i) * M + t * 16 + col] = acc[t][i];
}

// ---------------------------------------------------------------------------
// Per-node attention coefficients: al_s[n,h] = sum_c h[n,h,c]*a_src[h,c]
// ---------------------------------------------------------------------------
template <int H, int C>
__global__ void attn_coef(const float* __restrict__ h,
                          const float* __restrict__ a_src,
                          const float* __restrict__ a_dst,
                          float* __restrict__ al_s, float* __restrict__ al_d,
                          int N) {
    int idx = blockIdx.x * blockDim.x + threadIdx.x;
    if (idx >= N * H) return;
    int n = idx / H, hh = idx % H;            // H power of two -> shift/mask
    const float* hp = h + (size_t)n * H * C + hh * C;
    const float* as = a_src + hh * C;
    const float* ad = a_dst + hh * C;
    float ss = 0.f, sd = 0.f;
#pragma unroll
    for (int c = 0; c < C; c++) {
        float v = hp[c];
        ss += v * as[c];
        sd += v * ad[c];
    }
    al_s[idx] = ss;
    al_d[idx] = sd;
}

// ---------------------------------------------------------------------------
// Zero/neg-inf init for segment buffers (deterministic each launch)
// ---------------------------------------------------------------------------
__global__ void init_seg(unsigned* __restrict__ m, float* __restrict__ s, int nm,
                         float* __restrict__ acc, int nacc) {
    int i = blockIdx.x * blockDim.x + threadIdx.x;
    if (i < nm) { m[i] = 0x007FFFFFu; /* f2ord(-inf) */ s[i] = 0.f; }
    if (i < nacc) acc[i] = 0.f;
}

// ---------------------------------------------------------------------------
// Edge pass A: segment max of LeakyReLU(al_s[src]+al_d[dst]) over dst
// one thread per (edge, head); self-loops appended for e >= E
// ---------------------------------------------------------------------------
template <int H>
__global__ void edge_max(const int* __restrict__ src, const int* __restrict__ dst,
                         int E, int N, const float* __restrict__ al_s,
                         const float* __restrict__ al_d, unsigned* __restrict__ m) {
    int idx = blockIdx.x * blockDim.x + threadIdx.x;
    int tot = (E + N) * H;
    if (idx >= tot) return;
    int e = idx / H, hh = idx % H;            // H power of two -> shift/mask
    int s_ = (e < E) ? src[e] : (e - E);
    int d_ = (e < E) ? dst[e] : (e - E);
    float x = al_s[s_ * H + hh] + al_d[d_ * H + hh];
    x = (x >= 0.f) ? x : 0.2f * x;
    atomicMax(&m[d_ * H + hh], f2ord(x));
}

// ---------------------------------------------------------------------------
// Edge pass B (fused): w = exp(e - m[dst]); s[dst]+=w; acc[dst] += w*h[src]
// one wave per edge; lanes cover the H*C channels contiguously (coalesced
// gathers of h[src] and coalesced L2-resident atomic adds, no return value).
// Wave id forced scalar via readfirstlane -> src/dst become s_load and the
// segment-index arithmetic runs on the SALU.
// ---------------------------------------------------------------------------
template <int H, int C>
__global__ void edge_aggregate(const int* __restrict__ src,
                               const int* __restrict__ dst, int E, int N,
                               const float* __restrict__ al_s,
                               const float* __restrict__ al_d,
                               const unsigned* __restrict__ m,
                               const float* __restrict__ h,
                               float* __restrict__ s_sum,
                               float* __restrict__ acc) {
    constexpr int TC = H * C;   // 64 (layer1) or 32 (layer2)
    constexpr int PL = TC / 32; // elems per lane: 2 or 1
    int wave = (blockIdx.x * blockDim.x + threadIdx.x) >> 5;
    wave = __builtin_amdgcn_readfirstlane(wave);   // provably wave-uniform
    int lane = threadIdx.x & 31;
    if (wave >= E + N) return;
    int s_ = (wave < E) ? src[wave] : (wave - E);
    int d_ = (wave < E) ? dst[wave] : (wave - E);
    int c0 = lane * PL;
    int hh = c0 / C;
    float logit = al_s[s_ * H + hh] + al_d[d_ * H + hh];
    logit = (logit >= 0.f) ? logit : 0.2f * logit;
    float w = __expf(logit - ord2f(m[d_ * H + hh]));
    if ((c0 % C) == 0) atomicAdd(&s_sum[d_ * H + hh], w);
    const float* hs = h + (size_t)s_ * TC + c0;
    float* ap = acc + (size_t)d_ * TC + c0;
#pragma unroll
    for (int i = 0; i < PL; i++) atomicAdd(&ap[i], w * hs[i]);
}

// ---------------------------------------------------------------------------
// Layer-1 finalize (in place): out = acc/(s+1e-16) + b1 ; then ELU
// ---------------------------------------------------------------------------
template <int H, int C>
__global__ void finalize1(float* __restrict__ acc, const float* __restrict__ s_sum,
                          const float* __restrict__ b, int N) {
    int idx = blockIdx.x * blockDim.x + threadIdx.x;
    if (idx >= N * H * C) return;
    int n = idx / (H * C), hc = idx % (H * C), hh = hc / C;
    float v = acc[idx] / (s_sum[n * H + hh] + 1e-16f) + b[hc];
    acc[idx] = (v > 0.f) ? v : (__expf(v) - 1.f);
}

// ---------------------------------------------------------------------------
// Layer-2 finalize + log_softmax over 32 classes: one wave per node
// ---------------------------------------------------------------------------
__global__ void finalize2(const float* __restrict__ acc,
                          const float* __restrict__ s_sum,
                          const float* __restrict__ b,
                          float* __restrict__ out, int N) {
    int wave = (blockIdx.x * blockDim.x + threadIdx.x) >> 5;
    wave = __builtin_amdgcn_readfirstlane(wave);
    int lane = threadIdx.x & 31;
    if (wave >= N) return;
    float v = acc[(size_t)wave * 32 + lane] / (s_sum[wave] + 1e-16f) + b[lane];
    float mx = v;
#pragma unroll
    for (int o = 16; o > 0; o >>= 1) mx = fmaxf(mx, __shfl_xor(mx, o, 32));
    float ex = __expf(v - mx);
    float sm = ex;
#pragma unroll
    for (int o = 16; o > 0; o >>= 1) sm += __shfl_xor(sm, o, 32);
    out[(size_t)wave * 32 + lane] = v - mx - __logf(sm);
}

// ---------------------------------------------------------------------------
static inline int cdiv(long long a, int b) { return (int)((a + b - 1) / b); }

extern "C" void kernel_launch(void* const* d_in, const int* in_sizes, int n_in,
                              void* d_out, int out_size, void* d_ws, size_t ws_size,
                              hipStream_t stream) {
    const float* x     = (const float*)d_in[0];
    const int*   eidx  = (const int*)d_in[1];
    const float* W1    = (const float*)d_in[2];
    const float* aS1   = (const float*)d_in[3];
    const float* aD1   = (const float*)d_in[4];
    const float* b1    = (const float*)d_in[5];
    const float* W2    = (const float*)d_in[6];
    const float* aS2   = (const float*)d_in[7];
    const float* aD2   = (const float*)d_in[8];
    const float* b2    = (const float*)d_in[9];

    const int N = in_sizes[0] / 256;      // 100000
    const int E = in_sizes[1] / 2;        // 1600000
    const int* src = eidx;
    const int* dst = eidx + E;

    // Workspace carve-up (floats): 228*N floats ~ 91 MB
    float* w = (float*)d_ws;
    float*    h1   = w;              w += (size_t)64 * N;
    float*    alS1 = w;              w += (size_t)8 * N;
    float*    alD1 = w;              w += (size_t)8 * N;
    unsigned* m1   = (unsigned*)w;   w += (size_t)8 * N;
    float*    s1   = w;              w += (size_t)8 * N;
    float*    acc1 = w;              w += (size_t)64 * N;   // becomes ELU(h) in place
    float*    h2   = w;              w += (size_t)32 * N;
    float*    alS2 = w;              w += (size_t)1 * N;
    float*    alD2 = w;              w += (size_t)1 * N;
    unsigned* m2   = (unsigned*)w;   w += (size_t)1 * N;
    float*    s2   = w;              w += (size_t)1 * N;
    float*    acc2 = w;              w += (size_t)32 * N;
    (void)ws_size; (void)n_in; (void)out_size;

    const int BT = 256;

    // ---- Layer 1 ----
    gemm_wmma_f32<4><<<cdiv((long long)(N / 16) * 32, BT), BT, 0, stream>>>(x, W1, h1, N, 256);
    attn_coef<8, 8><<<cdiv((long long)N * 8, BT), BT, 0, stream>>>(h1, aS1, aD1, alS1, alD1, N);
    init_seg<<<cdiv((long long)64 * N, BT), BT, 0, stream>>>(m1, s1, 8 * N, acc1, 64 * N);
    edge_max<8><<<cdiv((long long)(E + N) * 8, BT), BT, 0, stream>>>(src, dst, E, N, alS1, alD1, m1);
    edge_aggregate<8, 8><<<cdiv((long long)(E + N) * 32, BT), BT, 0, stream>>>(
        src, dst, E, N, alS1, alD1, m1, h1, s1, acc1);
    finalize1<8, 8><<<cdiv((long long)64 * N, BT), BT, 0, stream>>>(acc1, s1, b1, N);

    // ---- Layer 2 ----
    gemm_wmma_f32<2><<<cdiv((long long)(N / 16) * 32, BT), BT, 0, stream>>>(acc1, W2, h2, N, 64);
    attn_coef<1, 32><<<cdiv((long long)N, BT), BT, 0, stream>>>(h2, aS2, aD2, alS2, alD2, N);
    init_seg<<<cdiv((long long)32 * N, BT), BT, 0, stream>>>(m2, s2, N, acc2, 32 * N);
    edge_max<1><<<cdiv((long long)(E + N), BT), BT, 0, stream>>>(src, dst, E, N, alS2, alD2, m2);
    edge_aggregate<1, 32><<<cdiv((long long)(E + N) * 32, BT), BT, 0, stream>>>(
        src, dst, E, N, alS2, alD2, m2, h2, s2, acc2);
    finalize2<<<cdiv((long long)N * 32, BT), BT, 0, stream>>>(acc2, s2, b2, (float*)d_out, N);
}